// QweNet_65438121721863
// MI455X (gfx1250) — compile-verified
//
#include <hip/hip_runtime.h>
#include <hip/hip_bf16.h>

typedef __attribute__((ext_vector_type(2))) float v2f;
typedef __attribute__((ext_vector_type(8))) float v8f;

#define DEPS 1e-5f

// ---------------- utility kernels ----------------

__global__ void k_zero(float* __restrict__ p, int n) {
    int i = blockIdx.x * 256 + threadIdx.x;
    if (i < n) p[i] = 0.0f;
}

__global__ void k_copyint(const int* __restrict__ a, int* __restrict__ b, int n) {
    int i = blockIdx.x * 256 + threadIdx.x;
    if (i < n) b[i] = a[i];
}

__global__ void k_deg(const int* __restrict__ dst, unsigned* __restrict__ deg, int E) {
    int e = blockIdx.x * 256 + threadIdx.x;
    if (e < E) atomicAdd(&deg[dst[e]], 1u);
}

__global__ void k_dinv(const unsigned* __restrict__ deg, float* __restrict__ dinv, int n) {
    int i = blockIdx.x * 256 + threadIdx.x;
    if (i < n) dinv[i] = rsqrtf((float)deg[i] + 1.0f);
}

// single-block exclusive scan over degree counts -> rowptr[N+1]
__global__ void k_scan(const unsigned* __restrict__ deg, int* __restrict__ rowptr, int N) {
    __shared__ int sh[1024];
    __shared__ int runningS;
    if (threadIdx.x == 0) runningS = 0;
    __syncthreads();
    for (int base = 0; base < N; base += 1024) {
        int i = base + (int)threadIdx.x;
        int v = (i < N) ? (int)deg[i] : 0;
        sh[threadIdx.x] = v;
        __syncthreads();
        for (int off = 1; off < 1024; off <<= 1) {
            int add = (threadIdx.x >= (unsigned)off) ? sh[threadIdx.x - off] : 0;
            __syncthreads();
            sh[threadIdx.x] += add;
            __syncthreads();
        }
        int incl = sh[threadIdx.x];
        int run = runningS;
        if (i < N) rowptr[i] = run + incl - v;  // exclusive
        __syncthreads();
        if (threadIdx.x == 1023) runningS = run + incl;
        __syncthreads();
    }
    if (threadIdx.x == 0) rowptr[N] = runningS;
}

// CSR fill: place each edge's src under its dst bucket
__global__ void k_fill(const int* __restrict__ src, const int* __restrict__ dst,
                       int* __restrict__ cursor, int* __restrict__ csr_src, int E) {
    int e = blockIdx.x * 256 + threadIdx.x;
    if (e < E) {
        int d = dst[e];
        int slot = atomicAdd(&cursor[d], 1);
        csr_src[slot] = src[e];
    }
}

// ---------------- fp32 WMMA GEMM: C[M x 128] = A[M x 128] @ W[128 x 128] (+ bias) ---------
// Block = 256 threads = 8 waves; blockIdx.x = 16-row stripe; wave w owns cols w*16..w*16+15.
// 32 x V_WMMA_F32_16X16X4_F32 per wave (K=128).

__global__ void k_gemm128(const float* __restrict__ A, const float* __restrict__ W,
                          const float* __restrict__ bias, float* __restrict__ C) {
    const int lane    = threadIdx.x & 31;
    const int wave    = threadIdx.x >> 5;
    const int rowBase = blockIdx.x << 4;
    const int colBase = wave << 4;
    const int r       = lane & 15;
    const int half    = lane >> 4;

    const float* __restrict__ Arow = A + (size_t)(rowBase + r) * 128;
    v8f acc = {};
#pragma unroll
    for (int k0 = 0; k0 < 128; k0 += 4) {
        const int ka = k0 + half * 2;
        v2f av, bv;
        av.x = Arow[ka];
        av.y = Arow[ka + 1];
        bv.x = W[ka * 128 + colBase + r];
        bv.y = W[(ka + 1) * 128 + colBase + r];
        acc = __builtin_amdgcn_wmma_f32_16x16x4_f32(false, av, false, bv,
                                                    (short)0, acc, false, false);
    }
    const int mb   = half * 8;
    const float bb = bias ? bias[colBase + r] : 0.0f;
#pragma unroll
    for (int i = 0; i < 8; ++i)
        C[(size_t)(rowBase + mb + i) * 128 + colBase + r] = acc[i] + bb;
}

// ---------------- GCN aggregation: CSR gather (no hot-loop atomics) ----------------
// 32 threads own one destination row; each thread covers 4 columns (float4).
// agg[n,:] = dinv[n]^2 * hw[n,:] + b + sum_{s in in(n)} dinv[n]*dinv[s] * hw[s,:]

__global__ void k_gather(const float* __restrict__ hw, const int* __restrict__ rowptr,
                         const int* __restrict__ csr_src, const float* __restrict__ dinv,
                         const float* __restrict__ b, float* __restrict__ agg, int N) {
    int t = blockIdx.x * 256 + threadIdx.x;  // N*32 threads
    if (t >= N * 32) return;
    int n = t >> 5;
    int part = (t & 31) << 2;

    float dn = dinv[n];
    float4 hv = *reinterpret_cast<const float4*>(hw + (size_t)n * 128 + part);
    float4 bv = *reinterpret_cast<const float4*>(b + part);
    float dn2 = dn * dn;
    float4 sum;
    sum.x = dn2 * hv.x + bv.x;
    sum.y = dn2 * hv.y + bv.y;
    sum.z = dn2 * hv.z + bv.z;
    sum.w = dn2 * hv.w + bv.w;

    int beg = rowptr[n], end = rowptr[n + 1];
    for (int j = beg; j < end; ++j) {
        int s = csr_src[j];                       // broadcast within the 32-lane group
        float coef = dn * dinv[s];
        float4 v = *reinterpret_cast<const float4*>(hw + (size_t)s * 128 + part);
        sum.x += coef * v.x;
        sum.y += coef * v.y;
        sum.z += coef * v.z;
        sum.w += coef * v.w;
    }
    *reinterpret_cast<float4*>(agg + (size_t)n * 128 + part) = sum;
}

// ---------------- column-wise batch stats (coalesced, atomic accumulate) ----------------
// Block covers all 128 cols (2 rows at a time -> contiguous 1KB loads).

__global__ void k_colsum(const float* __restrict__ z, float* __restrict__ sum,
                         float* __restrict__ sumsq, int M) {
    int col = threadIdx.x & 127;
    int sub = threadIdx.x >> 7;  // 0 or 1
    float s = 0.0f, s2 = 0.0f;
    for (int r = blockIdx.x * 2 + sub; r < M; r += gridDim.x * 2) {
        float v = z[(size_t)r * 128 + col];
        s += v;
        s2 += v * v;
    }
    __shared__ float sh[256], sh2[256];
    sh[threadIdx.x] = s;
    sh2[threadIdx.x] = s2;
    __syncthreads();
    if (sub == 0) {
        unsafeAtomicAdd(&sum[col], sh[col] + sh[col + 128]);
        unsafeAtomicAdd(&sumsq[col], sh2[col] + sh2[col + 128]);
    }
}

__global__ void k_stats_final(const float* __restrict__ sum, const float* __restrict__ sumsq,
                              float* __restrict__ mean, float* __restrict__ var, int M) {
    int c = threadIdx.x;  // 128 threads
    float m = sum[c] / (float)M;
    mean[c] = m;
    var[c] = sumsq[c] / (float)M - m * m;
}

// ---------------- fused BN + LeakyReLU (+ optional ||h||^2 accumulation) ----------------

__global__ void k_bn_lrelu(const float* __restrict__ z, const float* __restrict__ g,
                           const float* __restrict__ be, const float* __restrict__ mean,
                           const float* __restrict__ var, float* __restrict__ h,
                           float* norm2, int total) {
    int i = blockIdx.x * 256 + threadIdx.x;
    float sq = 0.0f;
    if (i < total) {
        int col = i & 127;
        float v = (z[i] - mean[col]) * rsqrtf(var[col] + DEPS) * g[col] + be[col];
        v = v > 0.0f ? v : 0.01f * v;
        h[i] = v;
        sq = v * v;
    }
    if (norm2) {
        __shared__ float sh[256];
        sh[threadIdx.x] = sq;
        __syncthreads();
        for (int off = 128; off > 0; off >>= 1) {
            if (threadIdx.x < (unsigned)off) sh[threadIdx.x] += sh[threadIdx.x + off];
            __syncthreads();
        }
        if (threadIdx.x == 0) unsafeAtomicAdd(norm2, sh[0]);
    }
}

__global__ void k_finalize_scale(float* __restrict__ norm2, float* __restrict__ scale) {
    scale[0] = rsqrtf(norm2[0]);  // 1/||h||_F
    norm2[0] = 0.0f;              // reset so each replay is identical
}

__global__ void k_scale(float* __restrict__ h, const float* __restrict__ scale, int total) {
    int i = blockIdx.x * 256 + threadIdx.x;
    if (i < total) h[i] *= scale[0];
}

// ---------------- decoder output: out = dh @ W_o + b_o (wave per row) ----------------

__global__ void k_decoder_out(const float* __restrict__ dh, const float* __restrict__ Wo,
                              const float* __restrict__ bo, float* __restrict__ out, int M) {
    int wave = threadIdx.x >> 5;
    int lane = threadIdx.x & 31;
    int r = blockIdx.x * 8 + wave;
    if (r >= M) return;
    float4 v = *reinterpret_cast<const float4*>(dh + (size_t)r * 128 + lane * 4);
    float4 w = *reinterpret_cast<const float4*>(Wo + lane * 4);
    float s = v.x * w.x + v.y * w.y + v.z * w.z + v.w * w.w;
#pragma unroll
    for (int off = 16; off > 0; off >>= 1) s += __shfl_down(s, off);
    if (lane == 0) out[r] = s + bo[0];
}

// ---------------- host-side orchestration ----------------

extern "C" void kernel_launch(void* const* d_in, const int* in_sizes, int n_in,
                              void* d_out, int out_size, void* d_ws, size_t ws_size,
                              hipStream_t stream) {
    (void)n_in; (void)out_size; (void)ws_size;
    const float* x     = (const float*)d_in[0];
    const int*   eidx  = (const int*)d_in[1];
    const float* W_in  = (const float*)d_in[2];
    const float* b_in  = (const float*)d_in[3];
    const float* g_in  = (const float*)d_in[4];
    const float* be_in = (const float*)d_in[5];
    const float* W_gcn = (const float*)d_in[6];
    const float* b_gcn = (const float*)d_in[7];
    const float* g_bn  = (const float*)d_in[8];
    const float* be_bn = (const float*)d_in[9];
    const float* W_h   = (const float*)d_in[10];
    const float* b_h   = (const float*)d_in[11];
    const float* g_d   = (const float*)d_in[12];
    const float* be_d  = (const float*)d_in[13];
    const float* W_o   = (const float*)d_in[14];
    const float* b_o   = (const float*)d_in[15];
    float* out = (float*)d_out;

    const int N = in_sizes[0] / 128;  // 50000
    const int E = in_sizes[1] / 2;    // 1600000
    const int total = N * 128;

    const int* src = eidx;
    const int* dst = eidx + E;

    // workspace layout
    float* hw      = (float*)d_ws;
    float* h       = hw + (size_t)total;
    float* agg     = h + (size_t)total;
    float* dinv    = agg + (size_t)total;
    unsigned* deg  = (unsigned*)(dinv + N);
    int* rowptr    = (int*)(deg + N);
    int* cursor    = rowptr + (N + 1);
    int* csr_src   = cursor + N;
    float* colsum  = (float*)(csr_src + E);
    float* colsumsq = colsum + 128;
    float* mean    = colsumsq + 128;
    float* var     = mean + 128;
    float* norm2   = var + 128;
    float* scale   = norm2 + 1;

    const int nbN   = (N + 255) / 256;
    const int nbE   = (E + 255) / 256;
    const int nbEl  = (total + 255) / 256;
    const int nbG   = (N * 32 + 255) / 256;
    const int nbGem = N / 16;  // N multiple of 16 (50000 = 3125*16)
    const int nbOut = (N + 7) / 8;

    // ---- CSR build (h-independent) + zero norm accumulator ----
    k_zero<<<nbN, 256, 0, stream>>>((float*)deg, N);
    k_zero<<<1, 256, 0, stream>>>(norm2, 1);
    k_deg<<<nbE, 256, 0, stream>>>(dst, deg, E);
    k_scan<<<1, 1024, 0, stream>>>(deg, rowptr, N);
    k_dinv<<<nbN, 256, 0, stream>>>(deg, dinv, N);
    k_copyint<<<nbN, 256, 0, stream>>>(rowptr, cursor, N);
    k_fill<<<nbE, 256, 0, stream>>>(src, dst, cursor, csr_src, E);

    // ---- encoder input layer: h = lrelu(BN(x@W_in + b_in)); h /= ||h|| ----
    k_gemm128<<<nbGem, 256, 0, stream>>>(x, W_in, b_in, hw);
    k_zero<<<1, 256, 0, stream>>>(colsum, 256);
    k_colsum<<<256, 256, 0, stream>>>(hw, colsum, colsumsq, N);
    k_stats_final<<<1, 128, 0, stream>>>(colsum, colsumsq, mean, var, N);
    k_bn_lrelu<<<nbEl, 256, 0, stream>>>(hw, g_in, be_in, mean, var, h, norm2, total);
    k_finalize_scale<<<1, 1, 0, stream>>>(norm2, scale);
    k_scale<<<nbEl, 256, 0, stream>>>(h, scale, total);

    // ---- 5x: h = lrelu(BN(gcn_conv(h))) ----
    for (int it = 0; it < 5; ++it) {
        k_gemm128<<<nbGem, 256, 0, stream>>>(h, W_gcn, nullptr, hw);
        k_gather<<<nbG, 256, 0, stream>>>(hw, rowptr, csr_src, dinv, b_gcn, agg, N);
        k_zero<<<1, 256, 0, stream>>>(colsum, 256);
        k_colsum<<<256, 256, 0, stream>>>(agg, colsum, colsumsq, N);
        k_stats_final<<<1, 128, 0, stream>>>(colsum, colsumsq, mean, var, N);
        k_bn_lrelu<<<nbEl, 256, 0, stream>>>(agg, g_bn, be_bn, mean, var, h,
                                             (it == 4) ? norm2 : nullptr, total);
    }
    k_finalize_scale<<<1, 1, 0, stream>>>(norm2, scale);
    k_scale<<<nbEl, 256, 0, stream>>>(h, scale, total);

    // ---- decoder ----
    k_gemm128<<<nbGem, 256, 0, stream>>>(h, W_h, b_h, hw);
    k_zero<<<1, 256, 0, stream>>>(colsum, 256);
    k_colsum<<<256, 256, 0, stream>>>(hw, colsum, colsumsq, N);
    k_stats_final<<<1, 128, 0, stream>>>(colsum, colsumsq, mean, var, N);
    k_bn_lrelu<<<nbEl, 256, 0, stream>>>(hw, g_d, be_d, mean, var, agg, nullptr, total);
    k_decoder_out<<<nbOut, 256, 0, stream>>>(agg, W_o, b_o, out, N);
}